// HashEmbedder6D_56315611185294
// MI455X (gfx1250) — compile-verified
//
#include <hip/hip_runtime.h>

// ---------------------------------------------------------------------------
// HashNeRF-style 6-D multiresolution hash embedding for MI455X (gfx1250).
// B points x 16 levels x 64 hypercube vertices; gather-bound in L2 (tables =
// 64MB total, fit in 192MB L2). Final per-point 64-way reduction is done on
// the matrix pipe via V_WMMA_F32_16X16X4_F32 using a diagonal formulation:
//   wave handles 16 points; lane p owns verts v%4 in {0,1}, lane p+16 owns
//   v%4 in {2,3} of point p  ->  WMMA A (wv) and B (emb) operands are formed
//   with ZERO cross-lane data movement per the 16x4 f32 operand layouts.
// ---------------------------------------------------------------------------

typedef float v2f __attribute__((ext_vector_type(2)));
typedef float v8f __attribute__((ext_vector_type(8)));

#define HASH_T      (1u << 19)
#define HASH_MASK   (HASH_T - 1u)
#define N_LEVELS    16
#define PTS_PER_BLK 128          // 8 waves * 16 points
#define THREADS     256

__device__ __constant__ float RES_TBL[N_LEVELS] = {
    16.f, 20.f, 25.f, 32.f, 40.f, 50.f, 64.f, 80.f,
    101.f, 128.f, 161.f, 203.f, 256.f, 322.f, 406.f, 512.f};

__device__ __constant__ unsigned PRIMES_D[6] = {
    1u, 2654435761u, 805459861u, 3674653429u, 2097192037u, 1434869437u};

__global__ __launch_bounds__(THREADS)
void hash6d_wmma_kernel(const float* __restrict__ x,
                        const float* __restrict__ tables,
                        float* __restrict__ out,
                        float* __restrict__ mask_out) {
    __shared__ __align__(16) float lds_out[8 * 16 * 32];   // 16 KB

    const int tid  = threadIdx.x;
    const int wave = tid >> 5;
    const int lane = tid & 31;
    const int hi   = lane >> 4;          // 0: verts r in {0,1}; 1: r in {2,3}
    const int pl   = lane & 15;          // point within wave tile
    const int blockBase = blockIdx.x * PTS_PER_BLK;
    const int pid = blockBase + wave * 16 + pl;

    // ---- load this lane's point (both half-lanes load the same point) ----
    const float* xp = x + (size_t)pid * 6;   // 24B stride, 8B aligned
    float2 p0 = *(const float2*)(xp + 0);
    float2 p1 = *(const float2*)(xp + 2);
    float2 p2 = *(const float2*)(xp + 4);
    float xd[6] = {p0.x, p0.y, p1.x, p1.y, p2.x, p2.y};

    float xc[6];
    bool keep = true;
#pragma unroll
    for (int d = 0; d < 6; ++d) {
        xc[d] = fminf(fmaxf(xd[d], -1.0f), 1.0f);
        keep = keep && (xd[d] == xc[d]);
    }

#pragma unroll 1
    for (int l = 0; l < N_LEVELS; ++l) {
        const float res = RES_TBL[l];
        const float g   = 2.0f / res;                 // grid (matches JAX)

        unsigned hbase = 0u;
        unsigned hx[6];
        float w[6], ow[6];
#pragma unroll
        for (int d = 0; d < 6; ++d) {
            float blf  = floorf((xc[d] + 1.0f) / g);  // (xc - box_min)/grid
            float vmin = blf * g - 1.0f;              // bl*grid + box_min
            float vmax = vmin + g;
            float wd   = (xd[d] - vmin) / ((vmax - vmin) + 1e-6f);
            wd = fminf(fmaxf(wd, 0.0f), 1.0f);
            int bl = (int)blf;
            unsigned h0 = (unsigned)bl * PRIMES_D[d];
            unsigned h1 = (unsigned)(bl + 1) * PRIMES_D[d];
            hbase ^= h0;
            hx[d] = h0 ^ h1;
            w[d]  = wd;
            ow[d] = 1.0f - wd;
        }

        // lane-private r-pair constants (r = vert index mod 4)
        const unsigned cr0 = hi ? hx[1] : 0u;             // r=2 : r=0
        const unsigned cr1 = hi ? (hx[0] ^ hx[1]) : hx[0];// r=3 : r=1
        const float wr0 = hi ? (ow[0] * w[1]) : (ow[0] * ow[1]);
        const float wr1 = hi ? (w[0]  * w[1]) : (w[0]  * ow[1]);

        const float w23[4] = {ow[2]*ow[3], w[2]*ow[3], ow[2]*w[3], w[2]*w[3]};
        const float w45[4] = {ow[4]*ow[5], w[4]*ow[5], ow[4]*w[5], w[4]*w[5]};
        const unsigned hx2 = hx[2], hx3 = hx[3], hx4 = hx[4], hx5 = hx[5];

        const float2* __restrict__ tl =
            (const float2*)(tables + (size_t)l * (size_t)HASH_T * 2);

        v8f acc0 = {};   // feature 0, 16x16 tile; diagonal = outputs
        v8f acc1 = {};   // feature 1

#pragma unroll
        for (int c = 0; c < 16; ++c) {                 // vert chunk 4c..4c+3
            unsigned hc = hbase;
            if (c & 1) hc ^= hx2;
            if (c & 2) hc ^= hx3;
            if (c & 4) hc ^= hx4;
            if (c & 8) hc ^= hx5;
            unsigned i0 = (hc ^ cr0) & HASH_MASK;
            unsigned i1 = (hc ^ cr1) & HASH_MASK;
            float2 e0 = tl[i0];                        // global_load_b64 gather
            float2 e1 = tl[i1];

            float wrest = w23[c & 3] * w45[c >> 2];
            v2f A;  A[0]  = wr0 * wrest;  A[1]  = wr1 * wrest;
            v2f B0; B0[0] = e0.x;         B0[1] = e1.x;
            v2f B1; B1[0] = e0.y;         B1[1] = e1.y;

            acc0 = __builtin_amdgcn_wmma_f32_16x16x4_f32(
                false, A, false, B0, (short)0, acc0, false, false);
            acc1 = __builtin_amdgcn_wmma_f32_16x16x4_f32(
                false, A, false, B1, (short)0, acc1, false, false);
        }

        // ---- extract diagonal D[p,p]: lanes 0-7 hold p=0..7 (vgpr=lane),
        //      lanes 24-31 hold p=8..15 (vgpr=lane-24) ----
        const bool vLo = (lane < 8);
        const bool vHi = (lane >= 24);
        const int  sel = vLo ? lane : (lane - 24);
        float o0 = acc0[0], o1 = acc1[0];
#pragma unroll
        for (int i = 1; i < 8; ++i) {
            if (sel == i) { o0 = acc0[i]; o1 = acc1[i]; }
        }
        if (vLo || vHi) {
            int point = vLo ? lane : (lane - 16);
            int base  = wave * 512 + point * 32 + l * 2;
            lds_out[base + 0] = o0;
            lds_out[base + 1] = o1;
        }
    }

    if (hi == 0) mask_out[pid] = keep ? 1.0f : 0.0f;

    __syncthreads();

    // ---- coalesced output: 128 points * 32 floats = 16KB contiguous ----
    float4* __restrict__ outv = (float4*)(out + (size_t)blockBase * 32);
    const float4* ldsv = (const float4*)lds_out;
#pragma unroll
    for (int i = 0; i < 4; ++i) {
        int idx = i * THREADS + tid;       // 1024 float4 total
        outv[idx] = ldsv[idx];
    }
}

extern "C" void kernel_launch(void* const* d_in, const int* in_sizes, int n_in,
                              void* d_out, int out_size, void* d_ws, size_t ws_size,
                              hipStream_t stream) {
    (void)n_in; (void)out_size; (void)d_ws; (void)ws_size;
    const float* x      = (const float*)d_in[0];
    const float* tables = (const float*)d_in[1];
    float* out = (float*)d_out;

    const int B = in_sizes[0] / 6;                 // 524288
    float* mask = out + (size_t)B * 32;            // tuple output #2

    dim3 grid(B / PTS_PER_BLK);                    // 4096
    dim3 block(THREADS);                           // 8 wave32
    hipLaunchKernelGGL(hash6d_wmma_kernel, grid, block, 0, stream,
                       x, tables, out, mask);
}